// SkillPolicy_73890617360506
// MI455X (gfx1250) — compile-verified
//
#include <hip/hip_runtime.h>
#include <hip/hip_bf16.h>
#include <math.h>

// ---------------------------------------------------------------------------
// SkillPolicy forward for MI455X (gfx1250).
// All GEMM-shaped einsums run on V_WMMA_F32_16X16X4_F32 (fp32 in/out, so we
// match the fp32 reference bit-for-bit in precision class).  One wave owns a
// 16x16 C tile; K advances 4 per WMMA.
// ---------------------------------------------------------------------------

typedef float v2f __attribute__((ext_vector_type(2)));
typedef float v8f __attribute__((ext_vector_type(8)));

__device__ __forceinline__ v8f wmma4(v2f a, v2f b, v8f c) {
  // (neg_a, A, neg_b, B, c_mod, C, reuse_a, reuse_b)
  return __builtin_amdgcn_wmma_f32_16x16x4_f32(false, a, false, b, (short)0, c,
                                               false, false);
}

// Problem constants
#define BB   16    // batch
#define NN   128   // agents
#define HH   256   // hidden
#define GD   64    // goal_dim
#define OD   128   // obs_dim
#define LD   64    // local_dim
#define NS   16    // num_skills
#define CD   (LD + GD + HH)  // 384

// ---------------------------------------------------------------------------
// Generic C[M,N] = A[M,K] @ W[N,K]^T + bias  (W row-major [N,K], like torch
// Linear).  One wave per 16x16 tile of C.
// ---------------------------------------------------------------------------
__global__ __launch_bounds__(256) void gemm_bias_k(const float* __restrict__ A,
                                                   const float* __restrict__ W,
                                                   const float* __restrict__ bias,
                                                   float* __restrict__ C,
                                                   int M, int N, int K) {
  const int gw = (blockIdx.x * blockDim.x + threadIdx.x) >> 5;
  const int tilesN = N >> 4;
  const int m0 = (gw / tilesN) << 4;
  const int n0 = (gw % tilesN) << 4;
  if (m0 >= M) return;  // whole-wave uniform: EXEC stays all-1s for WMMA

  const int lane = threadIdx.x & 31;
  const int half = lane >> 4;
  const int r    = lane & 15;
  const int kb   = half << 1;

  const float* Ap = A + (size_t)(m0 + r) * K;  // A row for this lane
  const float* Wp = W + (size_t)(n0 + r) * K;  // B[k][n] = W[n][k]

  v8f acc = {};
  for (int k = 0; k < K; k += 4) {
    __builtin_prefetch(Ap + k + 64, 0, 1);  // global_prefetch_b8 64 floats ahead
    v2f a, b;
    a.x = Ap[k + kb]; a.y = Ap[k + kb + 1];
    b.x = Wp[k + kb]; b.y = Wp[k + kb + 1];
    acc = wmma4(a, b, acc);
  }
  const float bv = bias[n0 + r];
#pragma unroll
  for (int i = 0; i < 8; ++i)
    C[(size_t)(m0 + i + 8 * half) * N + n0 + r] = acc[i] + bv;
}

// ---------------------------------------------------------------------------
// Fused attention per (batch, 16-row tile of N):
//   logits = Q Kb^T / 16 ; mask = sigmoid(logits + g1 - g2)
//   scores = logits + log(mask+1e-8) ; attn = softmax(scores)
//   info   = (attn*mask) @ V
// Block = 256 threads = 8 waves.  Wave w computes the 16x16 logits tile for
// agent columns [16w,16w+16), then 2 rows of softmax, then 2 h-tiles of info.
// ---------------------------------------------------------------------------
__global__ __launch_bounds__(256) void attn_fused_k(const float* __restrict__ Q,
                                                    const float* __restrict__ Kb,
                                                    const float* __restrict__ Vb,
                                                    const float* __restrict__ u1,
                                                    const float* __restrict__ u2,
                                                    float* __restrict__ mask_out,
                                                    float* __restrict__ attn_out,
                                                    float* __restrict__ info) {
  __shared__ float S[16][NN + 4];   // scores
  __shared__ float Mk[16][NN + 4];  // mask
  __shared__ float P[16][NN + 4];   // attn * mask

  const int b    = blockIdx.y;
  const int n0   = blockIdx.x << 4;
  const int tid  = threadIdx.x;
  const int wave = tid >> 5;
  const int lane = tid & 31;
  const int half = lane >> 4;
  const int r    = lane & 15;
  const int kb   = half << 1;

  // ---- Step 1: logits tile (rows n0..n0+15, cols m0..m0+15), WMMA over h ---
  const int m0 = wave << 4;
  const float* qrow = Q + (size_t)(b * NN + n0 + r) * HH;
  const float* krow = Kb + (size_t)(b * NN + m0 + r) * HH;  // B[k][m]=K[m][k]
  v8f acc = {};
  for (int k = 0; k < HH; k += 4) {
    v2f a, bv;
    a.x  = qrow[k + kb]; a.y  = qrow[k + kb + 1];
    bv.x = krow[k + kb]; bv.y = krow[k + kb + 1];
    acc = wmma4(a, bv, acc);
  }

  const float inv_scale = 1.0f / (16.0f + 1e-12f);  // sqrt(256)+eps
#pragma unroll
  for (int i = 0; i < 8; ++i) {
    const int row = i + 8 * half;       // n within tile
    const int col = m0 + r;             // m
    const float logit = acc[i] * inv_scale;
    const size_t gidx = (size_t)(b * NN + n0 + row) * NN + col;
    const float U1 = u1[gidx], U2 = u2[gidx];
    const float g1 = -__logf(-__logf(U1 + 1e-20f) + 1e-20f);
    const float g2 = -__logf(-__logf(U2 + 1e-20f) + 1e-20f);
    const float msk = 1.0f / (1.0f + __expf(-(logit + g1 - g2)));
    mask_out[gidx] = msk;
    Mk[row][col]   = msk;
    S[row][col]    = logit + __logf(msk + 1e-8f);
  }
  __syncthreads();

  // ---- Step 2: softmax over 128 cols; wave handles rows 2w, 2w+1 ----------
#pragma unroll
  for (int rr = 0; rr < 2; ++rr) {
    const int row = (wave << 1) + rr;
    float v0 = S[row][lane +  0], v1 = S[row][lane + 32];
    float v2 = S[row][lane + 64], v3 = S[row][lane + 96];
    float mx = fmaxf(fmaxf(v0, v1), fmaxf(v2, v3));
#pragma unroll
    for (int off = 16; off > 0; off >>= 1) mx = fmaxf(mx, __shfl_xor(mx, off));
    float e0 = __expf(v0 - mx), e1 = __expf(v1 - mx);
    float e2 = __expf(v2 - mx), e3 = __expf(v3 - mx);
    float sum = e0 + e1 + e2 + e3;
#pragma unroll
    for (int off = 16; off > 0; off >>= 1) sum += __shfl_xor(sum, off);
    const float inv = 1.0f / sum;
    float* arow = attn_out + (size_t)(b * NN + n0 + row) * NN;
    const float a0 = e0 * inv, a1 = e1 * inv, a2 = e2 * inv, a3 = e3 * inv;
    arow[lane +  0] = a0; arow[lane + 32] = a1;
    arow[lane + 64] = a2; arow[lane + 96] = a3;
    P[row][lane +  0] = a0 * Mk[row][lane +  0];
    P[row][lane + 32] = a1 * Mk[row][lane + 32];
    P[row][lane + 64] = a2 * Mk[row][lane + 64];
    P[row][lane + 96] = a3 * Mk[row][lane + 96];
  }
  __syncthreads();

  // ---- Step 3: info = P @ V ; wave handles h-tiles (wave) and (wave+8) ----
#pragma unroll
  for (int t = 0; t < 2; ++t) {
    const int h0 = (wave + (t << 3)) << 4;
    v8f ac = {};
    for (int k = 0; k < NN; k += 4) {
      v2f a, bv;
      a.x = P[r][k + kb]; a.y = P[r][k + kb + 1];
      const float* vp = Vb + (size_t)(b * NN + k + kb) * HH + h0 + r;
      bv.x = vp[0]; bv.y = vp[HH];
      ac = wmma4(a, bv, ac);
    }
#pragma unroll
    for (int i = 0; i < 8; ++i)
      info[(size_t)(b * NN + n0 + i + 8 * half) * HH + h0 + r] = ac[i];
  }
}

// ---------------------------------------------------------------------------
// Skill head: logit[m, s] = concat(agents_local|goals|out)[m,:] @ Ws[s,:] + bs
// M = 2048, N = 16 (single tile wide), K = 384 read piecewise (no concat buf).
// ---------------------------------------------------------------------------
__global__ __launch_bounds__(256) void skill_head_k(const float* __restrict__ al,
                                                    const float* __restrict__ gl,
                                                    const float* __restrict__ op,
                                                    const float* __restrict__ Ws,
                                                    const float* __restrict__ bs,
                                                    float* __restrict__ logit) {
  const int gw = (blockIdx.x * blockDim.x + threadIdx.x) >> 5;
  const int m0 = gw << 4;
  if (m0 >= BB * NN) return;
  const int lane = threadIdx.x & 31;
  const int half = lane >> 4;
  const int r    = lane & 15;
  const int kb   = half << 1;

  const int m = m0 + r;
  const float* alp = al + (size_t)m * LD;
  const float* glp = gl + (size_t)m * GD;
  const float* opp = op + (size_t)m * HH;
  const float* wsp = Ws + (size_t)r * CD;  // N=16 -> column r

  v8f acc = {};
  for (int k = 0; k < CD; k += 4) {
    const int kk = k + kb;  // segment boundaries (64,128) are multiples of 4
    v2f a, b;
    if (kk < LD)            { a.x = alp[kk];            a.y = alp[kk + 1]; }
    else if (kk < LD + GD)  { a.x = glp[kk - LD];       a.y = glp[kk - LD + 1]; }
    else                    { a.x = opp[kk - LD - GD];  a.y = opp[kk - LD - GD + 1]; }
    b.x = wsp[kk]; b.y = wsp[kk + 1];
    acc = wmma4(a, b, acc);
  }
  const float bv = bs[r];
#pragma unroll
  for (int i = 0; i < 8; ++i)
    logit[(size_t)(m0 + i + 8 * half) * NS + r] = acc[i] + bv;
}

// ---------------------------------------------------------------------------
extern "C" void kernel_launch(void* const* d_in, const int* in_sizes, int n_in,
                              void* d_out, int out_size, void* d_ws, size_t ws_size,
                              hipStream_t stream) {
  const float* goals        = (const float*)d_in[0];
  const float* agents       = (const float*)d_in[1];
  const float* agents_local = (const float*)d_in[2];
  const float* u1           = (const float*)d_in[3];
  const float* u2           = (const float*)d_in[4];
  const float* Wq = (const float*)d_in[5];
  const float* bq = (const float*)d_in[6];
  const float* Wk = (const float*)d_in[7];
  const float* bk = (const float*)d_in[8];
  const float* Wv = (const float*)d_in[9];
  const float* bv = (const float*)d_in[10];
  const float* Wo = (const float*)d_in[11];
  const float* bo = (const float*)d_in[12];
  const float* Ws = (const float*)d_in[13];
  const float* bs = (const float*)d_in[14];

  const int M = BB * NN;  // 2048
  float* ws   = (float*)d_ws;
  float* Qb   = ws;                    // 2048*256
  float* Kb   = ws + (size_t)M * HH;   // 2048*256
  float* Vb   = Kb + (size_t)M * HH;
  float* info = Vb + (size_t)M * HH;

  float* out_logit = (float*)d_out;                 // [16,128,16]
  float* out_mask  = out_logit + (size_t)M * NS;    // [16,128,128]
  float* out_attn  = out_mask + (size_t)M * NN;     // [16,128,128]
  float* out_out   = out_attn + (size_t)M * NN;     // [16,128,256]

  // Projections: waves = (M/16)*(H/16) = 2048 -> 256 blocks of 8 waves
  const int projBlocks = ((M / 16) * (HH / 16) * 32) / 256;
  gemm_bias_k<<<projBlocks, 256, 0, stream>>>(goals,  Wq, bq, Qb, M, HH, GD);
  gemm_bias_k<<<projBlocks, 256, 0, stream>>>(agents, Wk, bk, Kb, M, HH, OD);
  gemm_bias_k<<<projBlocks, 256, 0, stream>>>(agents, Wv, bv, Vb, M, HH, OD);

  // Fused masked attention: one block per (n-tile, batch)
  attn_fused_k<<<dim3(NN / 16, BB), 256, 0, stream>>>(Qb, Kb, Vb, u1, u2,
                                                      out_mask, out_attn, info);

  // out = info @ Wo^T + bo
  gemm_bias_k<<<projBlocks, 256, 0, stream>>>(info, Wo, bo, out_out, M, HH, HH);

  // skill head: 128 waves -> 16 blocks
  skill_head_k<<<(M / 16) * 32 / 256, 256, 0, stream>>>(agents_local, goals,
                                                        out_out, Ws, bs, out_logit);
}